// inductive_GCN_no_feat_16174846836923
// MI455X (gfx1250) — compile-verified
//
#include <hip/hip_runtime.h>

typedef float v2f __attribute__((ext_vector_type(2)));
typedef float v8f __attribute__((ext_vector_type(8)));

#define N_NODES  100000
#define N_EDGES  1600000
#define HDIM     128
#define RESIDUAL 1.0f

// ---------------- degree / normalization ----------------
__global__ __launch_bounds__(256) void deg_kernel(const int* __restrict__ dst,
                                                  float* __restrict__ deg) {
    int e = blockIdx.x * blockDim.x + threadIdx.x;
    if (e < N_EDGES) atomicAdd(&deg[dst[e]], 1.0f);
}

__global__ __launch_bounds__(256) void dinv_kernel(float* __restrict__ d) {
    int i = blockIdx.x * blockDim.x + threadIdx.x;
    if (i < N_NODES) d[i] = rsqrtf(d[i] + 1.0f);
}

// ---------------- t = h @ W  via V_WMMA_F32_16X16X4_F32 ----------------
// Block: 256 threads = 8 waves. Each block computes a 16-row slab of the
// output across all 128 columns; wave w owns column tile [16w, 16w+16).
// A slab (16x128 fp32) staged in LDS once per block (stride 132 = no bank
// conflicts on the per-lane row reads). W rows stream from global (hot in L2,
// shared by all 6250 blocks).
__global__ __launch_bounds__(256) void gemm_kernel(const float* __restrict__ h,
                                                   const float* __restrict__ W,
                                                   float* __restrict__ t) {
    __shared__ float Alds[16][132];
    const int rowbase = blockIdx.x * 16;
    const int tid = threadIdx.x;

    // cooperative load of the 16x128 A slab (8 floats per thread, coalesced)
    #pragma unroll
    for (int j = 0; j < 8; ++j) {
        int idx = tid + 256 * j;
        int r = idx >> 7;
        int c = idx & 127;
        Alds[r][c] = h[(size_t)(rowbase + r) * HDIM + c];
    }
    __syncthreads();

    const int wave = tid >> 5;      // 0..7 -> output column tile
    const int lane = tid & 31;
    const int half = lane >> 4;     // 0: K pair (0,1), 1: K pair (2,3)
    const int r16  = lane & 15;
    const int nCol = wave * 16 + r16;

    v8f acc = {};
    #pragma unroll
    for (int k0 = 0; k0 < HDIM; k0 += 4) {
        const int ka = k0 + 2 * half;
        v2f a, b;
        // A 16x4: lane r16 holds row M=r16, K = ka, ka+1
        a.x = Alds[r16][ka];
        a.y = Alds[r16][ka + 1];
        // B 4x16: lane r16 holds col N=nCol, K = ka, ka+1
        b.x = W[(size_t)ka * HDIM + nCol];
        b.y = W[(size_t)(ka + 1) * HDIM + nCol];
        acc = __builtin_amdgcn_wmma_f32_16x16x4_f32(
            /*neg_a=*/false, a, /*neg_b=*/false, b,
            /*c_mod=*/(short)0, acc, /*reuse_a=*/false, /*reuse_b=*/false);
    }

    // C/D 16x16 f32: VGPR i holds M=i (lanes 0-15) and M=i+8 (lanes 16-31)
    #pragma unroll
    for (int i = 0; i < 8; ++i) {
        int row = rowbase + i + 8 * half;
        t[(size_t)row * HDIM + nCol] = acc[i];
    }
}

// ---------------- edge scatter: agg[dst] += t[src] * dinv[s]*dinv[d] -------
// One wave per edge: 32 lanes x float4 = 128 columns, fully coalesced b128
// loads of the source row; edge index is wave-uniform so src/dst/dinv loads
// scalarize. fp32 atomics land in L2 (51 MB agg buffer fits in 192 MB L2).
__global__ __launch_bounds__(256) void scatter_kernel(const float* __restrict__ t,
                                                      const int* __restrict__ src,
                                                      const int* __restrict__ dst,
                                                      const float* __restrict__ dinv,
                                                      float* __restrict__ agg) {
    int gid  = blockIdx.x * blockDim.x + threadIdx.x;
    int e    = gid >> 5;            // wave-uniform
    int lane = gid & 31;
    if (e >= N_EDGES) return;
    int s = src[e];
    int d = dst[e];
    float norm = dinv[s] * dinv[d];
    const float4 v = *((const float4*)(t + (size_t)s * HDIM) + lane);
    float* ap = agg + (size_t)d * HDIM + lane * 4;
    atomicAdd(ap + 0, v.x * norm);
    atomicAdd(ap + 1, v.y * norm);
    atomicAdd(ap + 2, v.z * norm);
    atomicAdd(ap + 3, v.w * norm);
}

// ---------------- combine: out = [relu](agg + t*dinv^2 + b + R*ori) --------
__global__ __launch_bounds__(256) void combine_kernel(const float* __restrict__ agg,
                                                      const float* __restrict__ t,
                                                      const float* __restrict__ dinv,
                                                      const float* __restrict__ bias,
                                                      const float* __restrict__ ori,
                                                      float* __restrict__ out,
                                                      int apply_relu) {
    int q = blockIdx.x * blockDim.x + threadIdx.x;   // float4 index; N*32 total
    int node = q >> 5;
    int c4   = (q & 31) * 4;
    float di = dinv[node];
    float d2 = di * di;
    size_t base = (size_t)node * HDIM + c4;
    float4 a  = *(const float4*)(agg  + base);
    float4 tv = *(const float4*)(t    + base);
    float4 o  = *(const float4*)(ori  + base);
    float4 bv = *(const float4*)(bias + c4);
    float4 r;
    r.x = a.x + tv.x * d2 + bv.x + RESIDUAL * o.x;
    r.y = a.y + tv.y * d2 + bv.y + RESIDUAL * o.y;
    r.z = a.z + tv.z * d2 + bv.z + RESIDUAL * o.z;
    r.w = a.w + tv.w * d2 + bv.w + RESIDUAL * o.w;
    if (apply_relu) {
        r.x = fmaxf(r.x, 0.0f);
        r.y = fmaxf(r.y, 0.0f);
        r.z = fmaxf(r.z, 0.0f);
        r.w = fmaxf(r.w, 0.0f);
    }
    *(float4*)(out + base) = r;
}

extern "C" void kernel_launch(void* const* d_in, const int* in_sizes, int n_in,
                              void* d_out, int out_size, void* d_ws, size_t ws_size,
                              hipStream_t stream) {
    const float* x   = (const float*)d_in[0];
    const int*   src = (const int*)  d_in[1];
    const int*   dst = (const int*)  d_in[2];
    const float* W1  = (const float*)d_in[3];
    const float* b1  = (const float*)d_in[4];
    const float* W2  = (const float*)d_in[5];
    const float* b2  = (const float*)d_in[6];
    float* out = (float*)d_out;

    // workspace layout: dinv [N], t [N*H], agg [N*H]   (~103 MB)
    char* w = (char*)d_ws;
    const size_t dinv_bytes = (((size_t)N_NODES * 4) + 255) & ~(size_t)255;
    const size_t feat_bytes = (size_t)N_NODES * HDIM * 4;
    float* dinv = (float*)w;
    float* t    = (float*)(w + dinv_bytes);
    float* agg  = (float*)(w + dinv_bytes + feat_bytes);

    const int nb_edge    = N_EDGES / 256;            // 6250
    const int nb_node    = (N_NODES + 255) / 256;    // 391
    const int nb_gemm    = N_NODES / 16;             // 6250 (exact)
    const int nb_scatter = (N_EDGES * 32) / 256;     // 200000 (exact)
    const int nb_combine = (N_NODES * 32) / 256;     // 12500 (exact)

    // degree + normalization
    hipMemsetAsync(dinv, 0, (size_t)N_NODES * 4, stream);
    deg_kernel <<<nb_edge, 256, 0, stream>>>(dst, dinv);
    dinv_kernel<<<nb_node, 256, 0, stream>>>(dinv);

    // layer 1: h = relu(conv(x, W1, b1) + ori)   (h lives in d_out)
    gemm_kernel<<<nb_gemm, 256, 0, stream>>>(x, W1, t);
    hipMemsetAsync(agg, 0, feat_bytes, stream);
    scatter_kernel<<<nb_scatter, 256, 0, stream>>>(t, src, dst, dinv, agg);
    combine_kernel<<<nb_combine, 256, 0, stream>>>(agg, t, dinv, b1, x, out, 1);

    // layer 2: h = relu(conv(h, W2, b2) + ori)
    gemm_kernel<<<nb_gemm, 256, 0, stream>>>(out, W2, t);
    hipMemsetAsync(agg, 0, feat_bytes, stream);
    scatter_kernel<<<nb_scatter, 256, 0, stream>>>(t, src, dst, dinv, agg);
    combine_kernel<<<nb_combine, 256, 0, stream>>>(agg, t, dinv, b2, x, out, 1);

    // layer 3: out = conv(h, W2, b2) + ori       (no relu on final output)
    gemm_kernel<<<nb_gemm, 256, 0, stream>>>(out, W2, t);
    hipMemsetAsync(agg, 0, feat_bytes, stream);
    scatter_kernel<<<nb_scatter, 256, 0, stream>>>(t, src, dst, dinv, agg);
    combine_kernel<<<nb_combine, 256, 0, stream>>>(agg, t, dinv, b2, x, out, 0);
}